// JaggedTexture_22574348108027
// MI455X (gfx1250) — compile-verified
//
#include <hip/hip_runtime.h>

// Jagged bilinear texture sampling for MI455X (gfx1250).
// Memory-bound gather kernel: texture atlas (~77 MB) is L2-resident (192 MB L2);
// all streaming traffic (x, query_dims, out; ~128 MB) uses non-temporal cache
// hints (th:TH_LOAD_NT / th:TH_STORE_NT on gfx1250) so it cannot evict the
// texture working set from L2. Roofline: ~205 MB compulsory HBM @ 23.3 TB/s.

__global__ __launch_bounds__(256) void jagged_bilerp_kernel(
    const float* __restrict__ x,      // [nq][2] float32
    const int*   __restrict__ qd,     // [nq][3] int32: h, w, off
    const float* __restrict__ tex,    // [total][3] float32
    float*       __restrict__ out,    // [nq][3] float32
    int nq)
{
    int q = blockIdx.x * blockDim.x + threadIdx.x;
    if (q >= nq) return;

    // Streaming loads: non-temporal (do not pollute L2, texture stays resident).
    float x0 = __builtin_nontemporal_load(x + 2 * q + 0);
    float x1 = __builtin_nontemporal_load(x + 2 * q + 1);
    int h    = __builtin_nontemporal_load(qd + 3 * q + 0);
    int w    = __builtin_nontemporal_load(qd + 3 * q + 1);
    int off  = __builtin_nontemporal_load(qd + 3 * q + 2);

    // y = clip(x, 0, 1); fu = y0*h; fv = y1*w
    float u  = fminf(fmaxf(x0, 0.0f), 1.0f);
    float v  = fminf(fmaxf(x1, 0.0f), 1.0f);
    float fu = u * (float)h;
    float fv = v * (float)w;

    // i0 = clip(floor(fu), 0, h-1), etc.
    int i0 = (int)floorf(fu);
    int j0 = (int)floorf(fv);
    i0 = min(max(i0, 0), h - 1);
    j0 = min(max(j0, 0), w - 1);
    int i1 = min(i0 + 1, h - 1);
    int j1 = min(j0 + 1, w - 1);

    // du/dv clipped fractional weights (from clipped i0/j0, as in reference).
    float du = fminf(fmaxf(fu - (float)i0, 0.0f), 1.0f);
    float dv = fminf(fmaxf(fv - (float)j0, 0.0f), 1.0f);

    float w00 = (1.0f - du) * (1.0f - dv);
    float w01 = (1.0f - du) * dv;
    float w10 = du * (1.0f - dv);
    float w11 = du * dv;

    // Gathers: default (regular-temporal) policy -> cached, L2-resident texture.
    const float* __restrict__ t00 = tex + 3 * (size_t)(off + i0 * w + j0);
    const float* __restrict__ t01 = tex + 3 * (size_t)(off + i0 * w + j1);
    const float* __restrict__ t10 = tex + 3 * (size_t)(off + i1 * w + j0);
    const float* __restrict__ t11 = tex + 3 * (size_t)(off + i1 * w + j1);

    float a0 = t00[0], a1 = t00[1], a2 = t00[2];
    float b0 = t01[0], b1 = t01[1], b2 = t01[2];
    float c0 = t10[0], c1 = t10[1], c2 = t10[2];
    float d0 = t11[0], d1 = t11[1], d2 = t11[2];

    float r0 = a0 * w00 + b0 * w01 + c0 * w10 + d0 * w11;
    float r1 = a1 * w00 + b1 * w01 + c1 * w10 + d1 * w11;
    float r2 = a2 * w00 + b2 * w01 + c2 * w10 + d2 * w11;

    // Streaming store: non-temporal, merges to a single b96 store per lane.
    __builtin_nontemporal_store(r0, out + 3 * q + 0);
    __builtin_nontemporal_store(r1, out + 3 * q + 1);
    __builtin_nontemporal_store(r2, out + 3 * q + 2);
}

extern "C" void kernel_launch(void* const* d_in, const int* in_sizes, int n_in,
                              void* d_out, int out_size, void* d_ws, size_t ws_size,
                              hipStream_t stream) {
    const float* x   = (const float*)d_in[0];  // (NQ, 2) float32
    const int*   qd  = (const int*)d_in[1];    // (NQ, 3) int32
    const float* tex = (const float*)d_in[2];  // (total, 3) float32
    float*       out = (float*)d_out;          // (NQ, 3) float32

    int nq = in_sizes[0] / 2;
    const int block = 256;
    int grid = (nq + block - 1) / block;
    hipLaunchKernelGGL(jagged_bilerp_kernel, dim3(grid), dim3(block), 0, stream,
                       x, qd, tex, out, nq);
}